// MiniBatchKMeans_17188459119174
// MI455X (gfx1250) — compile-verified
//
#include <hip/hip_runtime.h>

#define B_N 65536
#define D_N 128
#define K_N 1024
#define COLLAPSE_TOL2 0.25f   // (0.5)^2

typedef __attribute__((ext_vector_type(2))) float v2f;
typedef __attribute__((ext_vector_type(8))) float v8f;

// ---- CDNA5 async-cnt wait (builtin if declared, else raw SOPP asm) ----
#if defined(__has_builtin) && __has_builtin(__builtin_amdgcn_s_wait_asynccnt)
#define WAIT_ASYNC(n) __builtin_amdgcn_s_wait_asynccnt(n)
#else
#define WAIT_ASYNC(n) asm volatile("s_wait_asynccnt %0" ::"n"(n) : "memory")
#endif

// Async global -> LDS copy of 16 bytes per lane (GLOBAL_LOAD_ASYNC_TO_LDS_B128,
// tracked by ASYNCcnt). lds_dst is the LDS byte address (flat addr low 32 bits).
__device__ __forceinline__ void async_copy_b128(uint32_t lds_dst, const void* gsrc) {
  asm volatile("global_load_async_to_lds_b128 %0, %1, off"
               :: "v"(lds_dst), "v"(gsrc)
               : "memory");
}

// ---------------------------------------------------------------------------
// Row L2-norms^2: out[i] = sum_d rows[i][d]^2
// ---------------------------------------------------------------------------
__global__ void norms_kernel(const float* __restrict__ rows, float* __restrict__ out, int n) {
  int i = blockIdx.x * blockDim.x + threadIdx.x;
  if (i >= n) return;
  const float* p = rows + (size_t)i * D_N;
  float s = 0.f;
  for (int d = 0; d < D_N; ++d) s = fmaf(p[d], p[d], s);
  out[i] = s;
}

// ---------------------------------------------------------------------------
// Fused distance GEMM via V_WMMA_F32_16X16X4_F32 with double-buffered
// async global->LDS staging of the 16-center column tile.
// MODE==0: per-row argmin over K + clamped min squared distance (assign, mind2)
// MODE==1: per-row max of clamped squared distance (farthest)
// Block = 256 threads = 8 waves; each wave owns a 16-row x 16-col WMMA tile,
// looping 64 column tiles. Each wave issues exactly 2 async ops per tile, so
// s_wait_asynccnt 2 after issuing tile ct+1 guarantees tile ct has landed
// (async loads complete in order).
// ---------------------------------------------------------------------------
template <int MODE>
__global__ __launch_bounds__(256) void dist_kernel(const float* __restrict__ batch,
                                                   const float* __restrict__ cent,
                                                   const float* __restrict__ cnorm,
                                                   int* __restrict__ assignv,
                                                   float* __restrict__ outval) {
  __shared__ __align__(16) float ldsB[2][16 * D_N];   // 2 x 8KB tiles

  const int lane = threadIdx.x & 31;
  const int wave = threadIdx.x >> 5;
  const int hlf = lane >> 4;        // 0 or 1
  const int nloc = lane & 15;       // column-within-tile / row-within-A-slice
  const int row_base = blockIdx.x * 128 + wave * 16;

  // LDS byte addresses (flat LDS aperture keeps the offset in addr[31:0])
  const uint32_t lds0 = (uint32_t)(uintptr_t)(&ldsB[0][0]);
  const uint32_t lds1 = (uint32_t)(uintptr_t)(&ldsB[1][0]);
  const uint32_t toff = (uint32_t)threadIdx.x * 16u;

  // ---- prologue: async-stage tile 0 into buffer 0 (8KB = 2 x B128/lane) ----
  {
    const char* s = (const char*)cent;
    async_copy_b128(lds0 + toff, s + toff);
    async_copy_b128(lds0 + 4096u + toff, s + 4096 + toff);
  }

  // ---- preload A slice: lane holds half of row (row_base + nloc) ----
  v2f a[32];
  const v2f* ap = (const v2f*)(batch + (size_t)(row_base + nloc) * D_N);
#pragma unroll
  for (int s = 0; s < 32; ++s) a[s] = ap[2 * s + hlf];

  // ---- per-row squared norm (combine both lane halves) ----
  float pn = 0.f;
#pragma unroll
  for (int s = 0; s < 32; ++s) pn = fmaf(a[s].x, a[s].x, fmaf(a[s].y, a[s].y, pn));
  float a2 = pn + __shfl_xor(pn, 16, 32);   // full norm of row (row_base+nloc)

  float bestv[8];
  int besti[8];
#pragma unroll
  for (int r = 0; r < 8; ++r) {
    bestv[r] = (MODE == 0) ? 3.4e38f : -1.f;
    besti[r] = 0;
  }

  for (int ct = 0; ct < 64; ++ct) {
    const int col_base = ct * 16;
    const int cur = ct & 1;

    if (ct + 1 < 64) {
      // async-stage next tile into the other buffer (protected by the
      // end-of-previous-iteration barrier), then wait for current tile.
      const char* s = (const char*)(cent + (size_t)(col_base + 16) * D_N);
      const uint32_t nb = cur ? lds0 : lds1;
      async_copy_b128(nb + toff, s + toff);
      async_copy_b128(nb + 4096u + toff, s + 4096 + toff);
      WAIT_ASYNC(2);          // <=2 outstanding -> tile ct complete (in-order)
    } else {
      WAIT_ASYNC(0);
    }
    __syncthreads();          // whole tile visible to all waves

    const float c2v = cnorm[col_base + nloc];

    v8f acc = {0.f, 0.f, 0.f, 0.f, 0.f, 0.f, 0.f, 0.f};
    const float* tile = cur ? &ldsB[1][0] : &ldsB[0][0];
    const v2f* bp = (const v2f*)(tile + nloc * D_N);
#pragma unroll
    for (int s = 0; s < 32; ++s) {
      v2f bf = bp[2 * s + hlf];
      acc = __builtin_amdgcn_wmma_f32_16x16x4_f32(false, a[s], false, bf,
                                                  (short)0, acc, false, false);
    }

#pragma unroll
    for (int r = 0; r < 8; ++r) {
      float a2m = __shfl(a2, r + 8 * hlf, 32);          // norm of row m=r+8*hlf
      float d2 = fmaxf(a2m + c2v - 2.f * acc[r], 0.f);
      if (MODE == 0) {
        if (d2 < bestv[r]) { bestv[r] = d2; besti[r] = col_base + nloc; }
      } else {
        bestv[r] = fmaxf(bestv[r], d2);
      }
    }
    __syncthreads();          // protect buffer before next iteration's overwrite
  }

  // ---- reduce across the 16 lanes that share each output row ----
#pragma unroll
  for (int r = 0; r < 8; ++r) {
    float v = bestv[r];
    int bi = besti[r];
#pragma unroll
    for (int off = 1; off < 16; off <<= 1) {
      float v2 = __shfl_xor(v, off, 32);
      int i2 = __shfl_xor(bi, off, 32);
      if (MODE == 0) {
        if (v2 < v || (v2 == v && i2 < bi)) { v = v2; bi = i2; }
      } else {
        v = fmaxf(v, v2);
      }
    }
    if (nloc == 0) {                       // lanes 0 (rows r) and 16 (rows r+8)
      int row = row_base + r + 8 * hlf;
      if (MODE == 0) { assignv[row] = bi; outval[row] = v; }
      else outval[row] = v;
    }
  }
}

// ---------------------------------------------------------------------------
// Deterministic per-cluster scatter: block k scans all assignments.
// ---------------------------------------------------------------------------
__global__ void scatter_kernel(const float* __restrict__ batch, const int* __restrict__ assignv,
                               float* __restrict__ counts_b, float* __restrict__ sums_b) {
  const int k = blockIdx.x;
  const int d = threadIdx.x;   // 128 threads
  float sum = 0.f;
  int cnt = 0;
  for (int i = 0; i < B_N; ++i) {
    if (assignv[i] == k) { ++cnt; sum += batch[(size_t)i * D_N + d]; }
  }
  sums_b[(size_t)k * D_N + d] = sum;
  if (d == 0) counts_b[k] = (float)cnt;
}

// ---------------------------------------------------------------------------
// Center update + empty-cluster replacement.
// ---------------------------------------------------------------------------
__global__ void update_kernel(const float* __restrict__ cent, const float* __restrict__ counts,
                              const float* __restrict__ counts_b, const float* __restrict__ sums_b,
                              const float* __restrict__ batch,
                              float* __restrict__ outC, float* __restrict__ outN) {
  int t = blockIdx.x * blockDim.x + threadIdx.x;    // K*D = 131072
  int k = t >> 7, d = t & 127;
  float cb = counts_b[k];
  float c0 = counts[k];
  bool empty = (c0 == 0.f) && (cb == 0.f);
  unsigned ridx = (2654435761u * (unsigned)k + 12345u) & (B_N - 1);
  float sb = empty ? batch[(size_t)ridx * D_N + d] : sums_b[t];
  float cbe = empty ? 1.f : cb;
  float nc = c0 + cbe;
  float cv = cent[t];
  float out = (cbe > 0.f) ? (cv * c0 + sb) / nc : cv;
  outC[t] = out;
  if (d == 0) outN[k] = nc;
}

__global__ void init_kernel(int* __restrict__ loser, int* __restrict__ topidx) {
  int t = blockIdx.x * blockDim.x + threadIdx.x;    // 2048
  if (t < K_N) loser[t] = 0;
  else topidx[t - K_N] = 0;
}

// ---------------------------------------------------------------------------
// Pairwise collapsed-center detection. Integer atomicOr => deterministic.
// ---------------------------------------------------------------------------
__global__ void pdist_kernel(const float* __restrict__ nc, const float* __restrict__ ncnt,
                             int* __restrict__ loser) {
  int t = blockIdx.x * blockDim.x + threadIdx.x;    // K*K
  int i = t >> 10, j = t & (K_N - 1);
  if (j <= i) return;
  const float* pa = nc + (size_t)i * D_N;
  const float* pb = nc + (size_t)j * D_N;
  float d2 = 0.f;
  for (int d = 0; d < D_N; ++d) {
    float df = pa[d] - pb[d];
    d2 = fmaf(df, df, d2);
  }
  if (d2 < COLLAPSE_TOL2) {
    if (ncnt[i] <= ncnt[j]) atomicOr(&loser[i], 1);
    else atomicOr(&loser[j], 1);
  }
}

// ---------------------------------------------------------------------------
// Exact stable descending rank: rank(i) = #{j : f_j > f_i} + #{j<i : f_j==f_i}.
// Ranks form a permutation -> topidx[0..K-1] fully written, deterministic.
// ---------------------------------------------------------------------------
__global__ void rank_kernel(const float* __restrict__ farthest, int* __restrict__ topidx) {
  __shared__ float chunk[256];
  int i = blockIdx.x * 256 + threadIdx.x;
  float fi = farthest[i];
  int rank = 0;
  for (int base = 0; base < B_N; base += 256) {
    __syncthreads();
    chunk[threadIdx.x] = farthest[base + threadIdx.x];
    __syncthreads();
    for (int t = 0; t < 256; ++t) {
      float fj = chunk[t];
      int j = base + t;
      rank += (fj > fi || (fj == fi && j < i)) ? 1 : 0;
    }
  }
  if (rank < K_N) topidx[rank] = i;
}

__global__ void loser_apply_kernel(const int* __restrict__ loser, const int* __restrict__ topidx,
                                   const float* __restrict__ batch,
                                   float* __restrict__ outC, float* __restrict__ outN) {
  int k = blockIdx.x;
  if (!loser[k]) return;                 // uniform per block
  __shared__ int rs;
  if (threadIdx.x == 0) {
    int r = 0;
    for (int j = 0; j < k; ++j) r += loser[j];
    rs = r;
  }
  __syncthreads();
  int idx = topidx[rs];
  outC[(size_t)k * D_N + threadIdx.x] = batch[(size_t)idx * D_N + threadIdx.x];
  if (threadIdx.x == 0) outN[k] = 0.f;
}

__global__ void inertia_kernel(const float* __restrict__ mind2, float* __restrict__ outI) {
  __shared__ float sh[256];
  float s = 0.f;
  for (int i = threadIdx.x; i < B_N; i += 256) s += mind2[i];
  sh[threadIdx.x] = s;
  __syncthreads();
  for (int off = 128; off > 0; off >>= 1) {
    if (threadIdx.x < off) sh[threadIdx.x] += sh[threadIdx.x + off];
    __syncthreads();
  }
  if (threadIdx.x == 0) *outI = sh[0] / (float)B_N;
}

// ---------------------------------------------------------------------------
extern "C" void kernel_launch(void* const* d_in, const int* in_sizes, int n_in,
                              void* d_out, int out_size, void* d_ws, size_t ws_size,
                              hipStream_t stream) {
  (void)in_sizes; (void)n_in; (void)out_size; (void)ws_size;
  const float* batch   = (const float*)d_in[0];   // [B, D]
  const float* centers = (const float*)d_in[1];   // [K, D]
  const float* counts  = (const float*)d_in[2];   // [K]

  float* outC = (float*)d_out;       // [K, D]
  float* outN = outC + K_N * D_N;    // [K]
  float* outI = outN + K_N;          // [1]

  // workspace layout (floats/ints), total ~1.3 MB
  float* c2       = (float*)d_ws;
  float* nc2      = c2 + K_N;
  float* counts_b = nc2 + K_N;
  float* sums_b   = counts_b + K_N;          // K*D
  float* mind2    = sums_b + K_N * D_N;      // B
  float* farthest = mind2 + B_N;             // B
  int*   assignv  = (int*)(farthest + B_N);  // B
  int*   loser    = assignv + B_N;           // K
  int*   topidx   = loser + K_N;             // K

  init_kernel<<<8, 256, 0, stream>>>(loser, topidx);
  norms_kernel<<<K_N / 256, 256, 0, stream>>>(centers, c2, K_N);
  dist_kernel<0><<<B_N / 128, 256, 0, stream>>>(batch, centers, c2, assignv, mind2);
  scatter_kernel<<<K_N, 128, 0, stream>>>(batch, assignv, counts_b, sums_b);
  update_kernel<<<(K_N * D_N) / 256, 256, 0, stream>>>(centers, counts, counts_b, sums_b,
                                                       batch, outC, outN);
  norms_kernel<<<K_N / 256, 256, 0, stream>>>(outC, nc2, K_N);
  pdist_kernel<<<(K_N * K_N) / 256, 256, 0, stream>>>(outC, outN, loser);
  dist_kernel<1><<<B_N / 128, 256, 0, stream>>>(batch, outC, nc2, nullptr, farthest);
  rank_kernel<<<B_N / 256, 256, 0, stream>>>(farthest, topidx);
  loser_apply_kernel<<<K_N, 128, 0, stream>>>(loser, topidx, batch, outC, outN);
  inertia_kernel<<<1, 256, 0, stream>>>(mind2, outI);
}